// GeometricMessagePassing_17188459119285
// MI455X (gfx1250) — compile-verified
//
#include <hip/hip_runtime.h>
#include <hip/hip_bf16.h>

typedef __attribute__((ext_vector_type(2))) float v2f;
typedef __attribute__((ext_vector_type(8))) float v8f;

#define Hdim   128
#define NHEAD  8
#define HDdim  16
#define EDdim  32
#define FFdim  512

// ---------------------------------------------------------------------------
// GEMM: Y[N,M] = X[N,K] @ W[K,M] + bias[M], exact fp32 via V_WMMA_F32_16X16X4_F32
// Block = 128 threads (4 waves). Each wave owns one 16x16 output tile.
// The 16xK X tile is staged once per block into LDS (row stride K+4 dwords:
// stride mod 64 == 4, so the 16 A-fragment lanes hit disjoint bank pairs).
// Requires N % 16 == 0 (10000 = 625*16), K % 16 == 0, M % 64 == 0.
// A fragment (16x4 f32): lanes 0-15 hold K={k0,k0+1}, lanes 16-31 K={k0+2,k0+3}.
// B fragment (4x16 f32): same K split, lane&15 selects column.
// D (16x16 f32, 8 VGPRs): VGPR i -> row i (lanes 0-15) / row i+8 (lanes 16-31).
// ---------------------------------------------------------------------------
__global__ void gemm_bias_f32_wmma(const float* __restrict__ X,
                                   const float* __restrict__ W,
                                   const float* __restrict__ bias,
                                   float* __restrict__ Y,
                                   int K, int M) {
  extern __shared__ float sh[];        // 16 * (K+4) floats
  const int lane    = threadIdx.x & 31;
  const int wave    = threadIdx.x >> 5;
  const int rowTile = blockIdx.x << 4;
  const int colTile = ((blockIdx.y << 2) + wave) << 4;
  const int r       = lane & 15;
  const int half    = lane >> 4;       // 0 -> K pair {0,1}; 1 -> K pair {2,3}
  const int ldss    = K + 4;

  // cooperative stage of the 16 x K tile of X (shared by all 4 waves)
  for (int idx = threadIdx.x; idx < (K << 4); idx += 128) {
    const int rr = idx / K;
    const int cc = idx - rr * K;
    sh[rr * ldss + cc] = X[(size_t)(rowTile + rr) * K + cc];
  }
  __syncthreads();

  v8f acc = {};
  const float* __restrict__ ap = sh + r * ldss + (half << 1);
  const float* __restrict__ wp = W + (size_t)(half << 1) * M + colTile + r;
  const size_t wstep = (size_t)4 * M;

  #pragma unroll 4
  for (int k0 = 0; k0 < K; k0 += 4) {
    if ((k0 & 15) == 0)                      // folds to one copy per unroll group
      __builtin_prefetch(wp + 4 * wstep, 0, 3);  // near-scope speculative prefetch
    const v2f a = *(const v2f*)ap;           // ds_load_b64, bank-conflict free
    v2f b;
    b.x = wp[0];
    b.y = wp[M];
    acc = __builtin_amdgcn_wmma_f32_16x16x4_f32(false, a, false, b, (short)0, acc,
                                                false, false);
    ap += 4;
    wp += wstep;
  }

  const float bb   = bias[colTile + r];
  const int   orow = rowTile + (half << 3);
  #pragma unroll
  for (int i = 0; i < 8; ++i)
    Y[(size_t)(orow + i) * M + colTile + r] = acc[i] + bb;
}

// ---------------------------------------------------------------------------
// Ordered-uint encoding so unsigned atomicMax == float max.
// ---------------------------------------------------------------------------
__device__ __forceinline__ unsigned f32_to_ord(float f) {
  unsigned u = __float_as_uint(f);
  return (u & 0x80000000u) ? ~u : (u | 0x80000000u);
}
__device__ __forceinline__ float ord_to_f32(unsigned k) {
  unsigned u = (k & 0x80000000u) ? (k ^ 0x80000000u) : ~k;
  return __uint_as_float(u);
}

// One thread per (edge, head): score = q[src,h].k[dst,h]/4 + edge_attr[e]@We[:,h]+be[h]
// Stores raw score; atomic segment-max into mx_enc[dst,h].
__global__ void edge_score_kernel(const long long* __restrict__ ei,
                                  const float* __restrict__ ea,
                                  const float* __restrict__ We,
                                  const float* __restrict__ be,
                                  const float* __restrict__ q,
                                  const float* __restrict__ k,
                                  float* __restrict__ scores,
                                  unsigned* __restrict__ mx_enc,
                                  int E_) {
  const int t = blockIdx.x * blockDim.x + threadIdx.x;
  if (t >= E_ * NHEAD) return;
  const int e   = t >> 3;
  const int h   = t & 7;
  const int src = (int)ei[e];
  const int dst = (int)ei[E_ + e];

  float bias = be[h];
  const float* __restrict__ ap = ea + (size_t)e * EDdim;
  #pragma unroll
  for (int d = 0; d < EDdim; ++d) bias += ap[d] * We[d * NHEAD + h];

  const float* __restrict__ qp = q + (size_t)src * Hdim + h * HDdim;
  const float* __restrict__ kp = k + (size_t)dst * Hdim + h * HDdim;
  float s = 0.f;
  #pragma unroll
  for (int d = 0; d < HDdim; ++d) s += qp[d] * kp[d];
  s = s * 0.25f + bias;                      // 1/sqrt(16)

  scores[t] = s;
  atomicMax(&mx_enc[dst * NHEAD + h], f32_to_ord(s));
}

// ex = exp(score - mx[dst,h]); accumulate denominators.
__global__ void edge_exp_kernel(const long long* __restrict__ ei,
                                float* __restrict__ scores,
                                const unsigned* __restrict__ mx_enc,
                                float* __restrict__ den,
                                int E_) {
  const int t = blockIdx.x * blockDim.x + threadIdx.x;
  if (t >= E_ * NHEAD) return;
  const int e   = t >> 3;
  const int h   = t & 7;
  const int dst = (int)ei[E_ + e];
  const float mx = ord_to_f32(mx_enc[dst * NHEAD + h]);
  const float ex = expf(scores[t] - mx);
  scores[t] = ex;
  atomicAdd(&den[dst * NHEAD + h], ex);
}

// agg[dst,h,:] += v[src,h,:] * ex/(den+1e-8)
__global__ void edge_agg_kernel(const long long* __restrict__ ei,
                                const float* __restrict__ ex,
                                const float* __restrict__ den,
                                const float* __restrict__ v,
                                float* __restrict__ agg,
                                int E_) {
  const int t = blockIdx.x * blockDim.x + threadIdx.x;
  if (t >= E_ * NHEAD) return;
  const int e   = t >> 3;
  const int h   = t & 7;
  const int src = (int)ei[e];
  const int dst = (int)ei[E_ + e];
  const float attn = ex[t] / (den[dst * NHEAD + h] + 1e-8f);
  const float* __restrict__ vp = v   + (size_t)src * Hdim + h * HDdim;
  float*       __restrict__ ap = agg + (size_t)dst * Hdim + h * HDdim;
  #pragma unroll
  for (int d = 0; d < HDdim; ++d) atomicAdd(&ap[d], vp[d] * attn);
}

// y = LayerNorm(x + r) * g + b, one 128-thread block per row (H = 128)
__global__ void add_ln_kernel(const float* __restrict__ x,
                              const float* __restrict__ rres,
                              const float* __restrict__ g,
                              const float* __restrict__ b,
                              float* __restrict__ y) {
  __shared__ float red[4];
  const int row = blockIdx.x;
  const int tid = threadIdx.x;
  float s = x[(size_t)row * Hdim + tid] + rres[(size_t)row * Hdim + tid];

  float ws = s;
  #pragma unroll
  for (int off = 16; off > 0; off >>= 1) ws += __shfl_down(ws, off, 32);
  if ((tid & 31) == 0) red[tid >> 5] = ws;
  __syncthreads();
  const float mean = (red[0] + red[1] + red[2] + red[3]) * (1.f / 128.f);
  __syncthreads();

  const float d = s - mean;
  float wv = d * d;
  #pragma unroll
  for (int off = 16; off > 0; off >>= 1) wv += __shfl_down(wv, off, 32);
  if ((tid & 31) == 0) red[tid >> 5] = wv;
  __syncthreads();
  const float var = (red[0] + red[1] + red[2] + red[3]) * (1.f / 128.f);

  y[(size_t)row * Hdim + tid] = d * rsqrtf(var + 1e-5f) * g[tid] + b[tid];
}

// exact GELU (approximate=False): 0.5*x*(1+erf(x/sqrt(2))), in place
__global__ void gelu_kernel(float* __restrict__ a, int n) {
  const int i = blockIdx.x * blockDim.x + threadIdx.x;
  if (i < n) {
    const float x = a[i];
    a[i] = 0.5f * x * (1.f + erff(x * 0.70710678118654752f));
  }
}

// ---------------------------------------------------------------------------
extern "C" void kernel_launch(void* const* d_in, const int* in_sizes, int n_in,
                              void* d_out, int out_size, void* d_ws, size_t ws_size,
                              hipStream_t stream) {
  const float*     x   = (const float*)d_in[0];
  const long long* ei  = (const long long*)d_in[1];
  const float*     ea  = (const float*)d_in[2];
  const float*     Wq  = (const float*)d_in[3];
  const float*     bq  = (const float*)d_in[4];
  const float*     Wk  = (const float*)d_in[5];
  const float*     bk  = (const float*)d_in[6];
  const float*     Wv  = (const float*)d_in[7];
  const float*     bv  = (const float*)d_in[8];
  const float*     We  = (const float*)d_in[9];
  const float*     be  = (const float*)d_in[10];
  const float*     Wo  = (const float*)d_in[11];
  const float*     bo  = (const float*)d_in[12];
  const float*     g1  = (const float*)d_in[13];
  const float*     be1 = (const float*)d_in[14];
  const float*     W1  = (const float*)d_in[15];
  const float*     b1  = (const float*)d_in[16];
  const float*     W2  = (const float*)d_in[17];
  const float*     b2  = (const float*)d_in[18];
  const float*     g2  = (const float*)d_in[19];
  const float*     be2 = (const float*)d_in[20];
  float* out = (float*)d_out;

  const int N_ = in_sizes[0] / Hdim;   // 10000 (multiple of 16)
  const int E_ = in_sizes[2] / EDdim;  // 320000

  // ---- carve workspace (floats) ----
  float* ws = (float*)d_ws;
  float*    q      = ws;                 ws += (size_t)N_ * Hdim;
  float*    k      = ws;                 ws += (size_t)N_ * Hdim;
  float*    v      = ws;                 ws += (size_t)N_ * Hdim;
  float*    scores = ws;                 ws += (size_t)E_ * NHEAD;
  unsigned* mx     = (unsigned*)ws;      ws += (size_t)N_ * NHEAD;   // zero-init region start
  float*    den    = ws;                 ws += (size_t)N_ * NHEAD;
  float*    agg    = ws;                 ws += (size_t)N_ * Hdim;    // zero-init region end
  float*    tmp    = ws;                 ws += (size_t)N_ * Hdim;    // attn out, later h2
  float*    x1     = ws;                 ws += (size_t)N_ * Hdim;
  float*    h1     = ws;                 ws += (size_t)N_ * FFdim;

  // zero mx/den/agg in one capturable memset (contiguous by construction)
  hipMemsetAsync(mx, 0, (size_t)(N_ * NHEAD * 2 + N_ * Hdim) * sizeof(float), stream);

  const dim3 blk(128);
  const dim3 gH(N_ / 16, Hdim / 64);    // M=128 GEMMs
  const dim3 gF(N_ / 16, FFdim / 64);   // M=512 GEMM
  const size_t shH = (size_t)16 * (Hdim + 4) * sizeof(float);   // K=128 stage
  const size_t shF = (size_t)16 * (FFdim + 4) * sizeof(float);  // K=512 stage

  // 1) Q/K/V projections (fp32 WMMA)
  gemm_bias_f32_wmma<<<gH, blk, shH, stream>>>(x, Wq, bq, q, Hdim, Hdim);
  gemm_bias_f32_wmma<<<gH, blk, shH, stream>>>(x, Wk, bk, k, Hdim, Hdim);
  gemm_bias_f32_wmma<<<gH, blk, shH, stream>>>(x, Wv, bv, v, Hdim, Hdim);

  // 2) edge scores + scatter softmax + aggregation
  const int eh      = E_ * NHEAD;
  const int eblocks = (eh + 255) / 256;
  edge_score_kernel<<<eblocks, 256, 0, stream>>>(ei, ea, We, be, q, k, scores, mx, E_);
  edge_exp_kernel  <<<eblocks, 256, 0, stream>>>(ei, scores, mx, den, E_);
  edge_agg_kernel  <<<eblocks, 256, 0, stream>>>(ei, scores, den, v, agg, E_);

  // 3) output projection + residual + LN1
  gemm_bias_f32_wmma<<<gH, blk, shH, stream>>>(agg, Wo, bo, tmp, Hdim, Hdim);
  add_ln_kernel<<<N_, 128, 0, stream>>>(x, tmp, g1, be1, x1);

  // 4) FFN: x1 @ W1 -> gelu -> @ W2, residual + LN2
  gemm_bias_f32_wmma<<<gF, blk, shH, stream>>>(x1, W1, b1, h1, Hdim, FFdim);
  const int ffn = N_ * FFdim;
  gelu_kernel<<<(ffn + 255) / 256, 256, 0, stream>>>(h1, ffn);
  gemm_bias_f32_wmma<<<gH, blk, shF, stream>>>(h1, W2, b2, tmp, FFdim, Hdim);
  add_ln_kernel<<<N_, 128, 0, stream>>>(x1, tmp, g2, be2, out);
}